// WGCN_73512660238652
// MI455X (gfx1250) — compile-verified
//
#include <hip/hip_runtime.h>

typedef __attribute__((ext_vector_type(2))) float v2f;
typedef __attribute__((ext_vector_type(8))) float v8f;

#define NN 50000
#define DD 128
#define EE 800000
#define STRIPES (NN / 16)          // 3125 (exact)
#define GBLK ((STRIPES + 7) / 8)   // 391 blocks of 8 waves

// ---------------------------------------------------------------------------
// deg[s] += 1 for each edge (segment count over src)
// ---------------------------------------------------------------------------
__global__ void WGCN_deg_kernel(const long long* __restrict__ src,
                                float* __restrict__ deg, int E) {
    int e = blockIdx.x * blockDim.x + threadIdx.x;
    if (e < E) {
        int s = (int)src[e];
        unsafeAtomicAdd(&deg[s], 1.0f);
    }
}

// ---------------------------------------------------------------------------
// msg[dst] += x[src]  — one wave32 per edge, float4 per lane (32*16B = 512B row)
// ---------------------------------------------------------------------------
__global__ void WGCN_scatter_kernel(const float* __restrict__ x,
                                    const long long* __restrict__ src,
                                    const long long* __restrict__ dst,
                                    float* __restrict__ msg, int E) {
    int wave = (int)((blockIdx.x * blockDim.x + threadIdx.x) >> 5);
    int lane = threadIdx.x & 31;
    if (wave < E) {
        int s = (int)src[wave];
        int d = (int)dst[wave];
        const float4 v = *(const float4*)(x + (size_t)s * DD + lane * 4);
        float* o = msg + (size_t)d * DD + lane * 4;
        unsafeAtomicAdd(o + 0, v.x);
        unsafeAtomicAdd(o + 1, v.y);
        unsafeAtomicAdd(o + 2, v.z);
        unsafeAtomicAdd(o + 3, v.w);
    }
}

// ---------------------------------------------------------------------------
// xout = relu((msg @ Wrel + brel + x @ Wroot) * deg)
// Wrel/Wroot staged in LDS (128 KB of gfx1250's 320 KB/WGP). One wave owns a
// full 16x128 row stripe: 8 f32 accumulators, A fragments reused across all 8
// column tiles, B fragments fed from ds_load. 512 v_wmma_f32_16x16x4_f32/wave.
// Fragment layouts per ISA 7.12.2: A lane<16 -> K=k,k+1 ; lane>=16 -> K=k+2,k+3.
// C/D: VGPR j, upper half-wave = M+8.
// ---------------------------------------------------------------------------
__global__ __launch_bounds__(256) void WGCN_layer_gemm(
    const float* __restrict__ msg, const float* __restrict__ x,
    const float* __restrict__ Wrel, const float* __restrict__ brel,
    const float* __restrict__ Wroot, const float* __restrict__ deg,
    float* __restrict__ xout) {
    extern __shared__ float smem[];
    float* sR = smem;            // Wrel   128*128 f32
    float* sT = smem + DD * DD;  // Wroot  128*128 f32

    {   // cooperative 128 KB weight stage, float4 granularity
        const float4* gR  = (const float4*)Wrel;
        const float4* gT  = (const float4*)Wroot;
        float4*       s4R = (float4*)sR;
        float4*       s4T = (float4*)sT;
        for (int i = threadIdx.x; i < DD * DD / 4; i += 256) {
            s4R[i] = gR[i];
            s4T[i] = gT[i];
        }
    }
    __syncthreads();

    const int lane   = threadIdx.x & 31;
    const int wavei  = threadIdx.x >> 5;
    const int stripe = blockIdx.x * 8 + wavei;
    if (stripe >= STRIPES) return;   // uniform per wave: EXEC stays all-1s

    const int row0  = stripe * 16;
    const int mrow  = row0 + (lane & 15);
    const int khalf = (lane >> 4) * 2;   // 0 or 2
    const int nsub  = lane & 15;

    v8f acc[8];
#pragma unroll
    for (int t = 0; t < 8; ++t) acc[t] = 0.0f;

    for (int k = 0; k < DD; k += 4) {
        const int ka = k + khalf;
        v2f aM, aX;
        { float2 t2 = *(const float2*)(msg + (size_t)mrow * DD + ka); aM.x = t2.x; aM.y = t2.y; }
        { float2 t2 = *(const float2*)(x   + (size_t)mrow * DD + ka); aX.x = t2.x; aX.y = t2.y; }
#pragma unroll
        for (int t = 0; t < 8; ++t) {
            const int n = t * 16 + nsub;
            v2f bR, bT;
            bR.x = sR[ka * DD + n];
            bR.y = sR[(ka + 1) * DD + n];
            bT.x = sT[ka * DD + n];
            bT.y = sT[(ka + 1) * DD + n];
            acc[t] = __builtin_amdgcn_wmma_f32_16x16x4_f32(false, aM, false, bR,
                                                           (short)0, acc[t], false, false);
            acc[t] = __builtin_amdgcn_wmma_f32_16x16x4_f32(false, aX, false, bT,
                                                           (short)0, acc[t], false, false);
        }
    }

    const int mbase = row0 + ((lane >> 4) << 3);
    float degv[8];
#pragma unroll
    for (int j = 0; j < 8; ++j) degv[j] = deg[mbase + j];

#pragma unroll
    for (int t = 0; t < 8; ++t) {
        const int   n    = t * 16 + nsub;
        const float bias = brel[n];
#pragma unroll
        for (int j = 0; j < 8; ++j) {
            float v = (acc[t][j] + bias) * degv[j];
            xout[(size_t)(mbase + j) * DD + n] = v > 0.0f ? v : 0.0f;
        }
    }
}

// ---------------------------------------------------------------------------
// out = x @ Wlin + blin  (Wlin staged in 64 KB LDS, same stripe structure)
// ---------------------------------------------------------------------------
__global__ __launch_bounds__(256) void WGCN_final_gemm(
    const float* __restrict__ x, const float* __restrict__ Wlin,
    const float* __restrict__ blin, float* __restrict__ out) {
    extern __shared__ float smem[];
    float* sW = smem;

    {
        const float4* gW  = (const float4*)Wlin;
        float4*       s4W = (float4*)sW;
        for (int i = threadIdx.x; i < DD * DD / 4; i += 256) s4W[i] = gW[i];
    }
    __syncthreads();

    const int lane   = threadIdx.x & 31;
    const int wavei  = threadIdx.x >> 5;
    const int stripe = blockIdx.x * 8 + wavei;
    if (stripe >= STRIPES) return;

    const int row0  = stripe * 16;
    const int mrow  = row0 + (lane & 15);
    const int khalf = (lane >> 4) * 2;
    const int nsub  = lane & 15;

    v8f acc[8];
#pragma unroll
    for (int t = 0; t < 8; ++t) acc[t] = 0.0f;

    for (int k = 0; k < DD; k += 4) {
        const int ka = k + khalf;
        v2f a;
        { float2 t2 = *(const float2*)(x + (size_t)mrow * DD + ka); a.x = t2.x; a.y = t2.y; }
#pragma unroll
        for (int t = 0; t < 8; ++t) {
            const int n = t * 16 + nsub;
            v2f b;
            b.x = sW[ka * DD + n];
            b.y = sW[(ka + 1) * DD + n];
            acc[t] = __builtin_amdgcn_wmma_f32_16x16x4_f32(false, a, false, b,
                                                           (short)0, acc[t], false, false);
        }
    }

    const int mbase = row0 + ((lane >> 4) << 3);
#pragma unroll
    for (int t = 0; t < 8; ++t) {
        const int   n    = t * 16 + nsub;
        const float bias = blin[n];
#pragma unroll
        for (int j = 0; j < 8; ++j)
            out[(size_t)(mbase + j) * DD + n] = acc[t][j] + bias;
    }
}

// ---------------------------------------------------------------------------
extern "C" void kernel_launch(void* const* d_in, const int* in_sizes, int n_in,
                              void* d_out, int out_size, void* d_ws, size_t ws_size,
                              hipStream_t stream) {
    const float*     x0    = (const float*)d_in[0];
    const long long* ei    = (const long long*)d_in[1];
    const float*     Wrel  = (const float*)d_in[2];
    const float*     brel  = (const float*)d_in[3];
    const float*     Wroot = (const float*)d_in[4];
    const float*     Wlin  = (const float*)d_in[5];
    const float*     blin  = (const float*)d_in[6];
    float*           out   = (float*)d_out;

    const long long* src = ei;        // edge_index[0]
    const long long* dst = ei + EE;   // edge_index[1]

    // Workspace layout (floats): deg | msg | xb0  (~51 MB); d_out doubles as
    // a second ping-pong buffer between layers.
    float* deg = (float*)d_ws;
    float* msg = deg + ((NN + 63) & ~63);
    float* xb0 = msg + (size_t)NN * DD;

    hipMemsetAsync(deg, 0, sizeof(float) * NN, stream);
    WGCN_deg_kernel<<<(EE + 255) / 256, 256, 0, stream>>>(src, deg, EE);

    const int scatterGrid = (int)(((size_t)EE * 32 + 255) / 256);  // 100000 blocks
    const size_t ldsLayer = (size_t)2 * DD * DD * sizeof(float);   // 128 KB
    const size_t ldsFinal = (size_t)DD * DD * sizeof(float);       // 64 KB

    // Layer ping-pong: x0 -> xb0 -> out(temp) -> xb0 ; final: xb0 -> out
    const float* cur = x0;
    float*       nxt = xb0;
    for (int l = 0; l < 3; ++l) {
        hipMemsetAsync(msg, 0, sizeof(float) * (size_t)NN * DD, stream);
        WGCN_scatter_kernel<<<scatterGrid, 256, 0, stream>>>(cur, src, dst, msg, EE);
        WGCN_layer_gemm<<<GBLK, 256, ldsLayer, stream>>>(
            msg, cur,
            Wrel + (size_t)l * DD * DD, brel + (size_t)l * DD,
            Wroot + (size_t)l * DD * DD, deg, nxt);
        cur = nxt;
        nxt = (l == 0) ? out : xb0;   // L0->xb0, L1->out(temp), L2->xb0
    }
    WGCN_final_gemm<<<GBLK, 256, ldsFinal, stream>>>(cur, Wlin, blin, out);
}